// TransMIL_31808527794680
// MI455X (gfx1250) — compile-verified
//
#include <hip/hip_runtime.h>
#include <math.h>

// ---------------- model dims ----------------
#define NT      20479           // input tiles
#define NTOK    20480           // tokens incl CLS
#define INDIM   1024
#define DIM     512
#define HEADS   8
#define HD      64
#define LM      256             // landmarks
#define CHUNK   80              // NTOK / LM
#define MLPD    512
#define QSCALE  0.125f          // HD^-0.5

typedef __attribute__((ext_vector_type(2))) float v2f;
typedef __attribute__((ext_vector_type(4))) float v4f;
typedef __attribute__((ext_vector_type(8))) float v8f;

// =====================================================================
// Generic strided, batched fp32 GEMM using V_WMMA_F32_16X16X4_F32.
// C[b] = alpha * A[b] @ B[b]  (+ bias[n]) (gelu) (+ res[m,n])
// Preconditions: K % 16 == 0, N % 64 == 0 (M guarded).
// Fast paths: as_k==1 (A k-contiguous)  -> b128 loads
//             bs_n==1 or bs_k==1 (B contiguous) -> b128 loads
// LDS layouts:
//   As[r][k]          stride 20 -> 16B-aligned quads; A frag = one b64 load
//   Bsp[k/2][2n+k&1]  paired even/odd k -> B frag = one b64 load, no shuffles
// Block: 128 threads (4 waves). 64x64 C tile; each wave 16x64.
// Register-prefetch pipeline overlaps next tile's global loads with WMMA.
// =====================================================================
__global__ __launch_bounds__(128)
void gemm_wmma_f32(const float* __restrict__ A, const float* __restrict__ B,
                   float* __restrict__ C,
                   int M, int N, int K,
                   long long as_m, long long as_k, long long ab,
                   long long bs_k, long long bs_n, long long bb,
                   long long cs_m, long long cb,
                   const float* __restrict__ bias,
                   const float* __restrict__ res, long long rs_m,
                   float alpha, int act_gelu)
{
    __shared__ __align__(16) float As[64][20];    // A tile, stride 20
    __shared__ __align__(16) float Bsp[8][136];   // B tile, paired k layout, stride 136
    const int tid  = threadIdx.x;
    const int wave = tid >> 5;
    const int lane = tid & 31;
    const int half = lane >> 4;
    const int l16  = lane & 15;
    const long long row0 = (long long)blockIdx.y * 64;
    const long long col0 = (long long)blockIdx.x * 64;
    A += (long long)blockIdx.z * ab;
    B += (long long)blockIdx.z * bb;
    C += (long long)blockIdx.z * cb;

    const bool fastA = (as_k == 1);
    const int  bmode = (bs_n == 1) ? 0 : ((bs_k == 1) ? 1 : 2);

    float ra[8], rb[8];

    auto loadA = [&](int kb) {
        if (fastA) {
#pragma unroll
            for (int i = 0; i < 2; ++i) {
                int e = tid + i * 128;          // 256 quads: 64 rows x 4 k-quads
                int r = e >> 2, kq = (e & 3) << 2;
                long long gm = row0 + r;
                v4f v = {0.0f, 0.0f, 0.0f, 0.0f};
                if (gm < (long long)M)
                    v = *(const v4f*)(A + gm * as_m + (long long)(kb + kq));
                ra[i*4+0] = v.x; ra[i*4+1] = v.y; ra[i*4+2] = v.z; ra[i*4+3] = v.w;
            }
        } else {
#pragma unroll
            for (int i = 0; i < 8; ++i) {
                int e = tid + i * 128;
                int r = e >> 4, kk = e & 15;
                long long gm = row0 + r;
                ra[i] = (gm < (long long)M)
                      ? A[gm * as_m + (long long)(kb + kk) * as_k] : 0.0f;
            }
        }
    };
    auto storeA = [&]() {
        if (fastA) {
#pragma unroll
            for (int i = 0; i < 2; ++i) {
                int e = tid + i * 128;
                int r = e >> 2, kq = (e & 3) << 2;
                v4f v = {ra[i*4+0], ra[i*4+1], ra[i*4+2], ra[i*4+3]};
                *(v4f*)&As[r][kq] = v;
            }
        } else {
#pragma unroll
            for (int i = 0; i < 8; ++i) {
                int e = tid + i * 128;
                As[e >> 4][e & 15] = ra[i];
            }
        }
    };
    auto loadB = [&](int kb) {
        if (bmode == 0) {                       // bs_n == 1: quads along n
#pragma unroll
            for (int i = 0; i < 2; ++i) {
                int e = tid + i * 128;          // 16 rows x 16 n-quads
                int kk = e >> 4, nq = (e & 15) << 2;
                v4f v = *(const v4f*)(B + (long long)(kb + kk) * bs_k + col0 + nq);
                rb[i*4+0] = v.x; rb[i*4+1] = v.y; rb[i*4+2] = v.z; rb[i*4+3] = v.w;
            }
        } else if (bmode == 1) {                // bs_k == 1: quads along k
#pragma unroll
            for (int i = 0; i < 2; ++i) {
                int e = tid + i * 128;          // 64 cols x 4 k-quads
                int n = e >> 2, kq = (e & 3) << 2;
                v4f v = *(const v4f*)(B + (long long)(kb + kq) + (col0 + n) * bs_n);
                rb[i*4+0] = v.x; rb[i*4+1] = v.y; rb[i*4+2] = v.z; rb[i*4+3] = v.w;
            }
        } else {
#pragma unroll
            for (int i = 0; i < 8; ++i) {
                int e = tid + i * 128;
                int kk = e >> 6, n = e & 63;
                rb[i] = B[(long long)(kb + kk) * bs_k + (col0 + n) * bs_n];
            }
        }
    };
    auto storeB = [&]() {
        if (bmode == 0) {                       // row kk, 4 consecutive n
#pragma unroll
            for (int i = 0; i < 2; ++i) {
                int e = tid + i * 128;
                int kk = e >> 4, nq = (e & 15) << 2;
                int p = kk >> 1, c = kk & 1;
#pragma unroll
                for (int j = 0; j < 4; ++j) Bsp[p][2 * (nq + j) + c] = rb[i*4+j];
            }
        } else if (bmode == 1) {                // col n, 4 consecutive k: two pairs
#pragma unroll
            for (int i = 0; i < 2; ++i) {
                int e = tid + i * 128;
                int n = e >> 2, kq = (e & 3) << 2;
                v2f p0 = {rb[i*4+0], rb[i*4+1]};
                v2f p1 = {rb[i*4+2], rb[i*4+3]};
                *(v2f*)&Bsp[(kq >> 1) + 0][2 * n] = p0;
                *(v2f*)&Bsp[(kq >> 1) + 1][2 * n] = p1;
            }
        } else {
#pragma unroll
            for (int i = 0; i < 8; ++i) {
                int e = tid + i * 128;
                int kk = e >> 6, n = e & 63;
                Bsp[kk >> 1][2 * n + (kk & 1)] = rb[i];
            }
        }
    };

    v8f acc[4];
#pragma unroll
    for (int t = 0; t < 4; ++t)
#pragma unroll
        for (int v = 0; v < 8; ++v) acc[t][v] = 0.0f;

    loadA(0);
    loadB(0);
    for (int k0 = 0; k0 < K; k0 += 16) {
        storeA();
        storeB();
        __syncthreads();
        const int kn = k0 + 16;
        if (kn < K) { loadA(kn); loadB(kn); }   // overlap with WMMA below
#pragma unroll
        for (int kk = 0; kk < 16; kk += 4) {
            v2f a = *(const v2f*)&As[wave * 16 + l16][kk + 2 * half];
#pragma unroll
            for (int t = 0; t < 4; ++t) {
                v2f b = *(const v2f*)&Bsp[(kk >> 1) + half][2 * (t * 16 + l16)];
                acc[t] = __builtin_amdgcn_wmma_f32_16x16x4_f32(
                    false, a, false, b, (short)0, acc[t], false, false);
            }
        }
        __syncthreads();
    }

#pragma unroll
    for (int t = 0; t < 4; ++t) {
#pragma unroll
        for (int v = 0; v < 8; ++v) {
            long long m = row0 + wave * 16 + v + 8 * half;
            long long n = col0 + t * 16 + l16;
            if (m < (long long)M) {
                float val = alpha * acc[t][v];
                if (bias) val += bias[n];
                if (act_gelu) val = 0.5f * val * (1.0f + erff(val * 0.7071067811865475f));
                if (res) val += res[m * rs_m + n];
                C[m * cs_m + n] = val;
            }
        }
    }
}

// ---------------- morphology position encoding: PE[NT][512] ----------------
__global__ void pe_kernel(const int* __restrict__ coords, float* __restrict__ PE)
{
    long long idx = (long long)blockIdx.x * blockDim.x + threadIdx.x;
    if (idx >= (long long)NT * 128) return;
    int t = (int)(idx >> 7);
    int j = (int)(idx & 127);
    float div = expf(-logf(10000.0f) * (float)j / 256.0f);
    float x = (float)coords[2 * t];
    float y = (float)coords[2 * t + 1];
    float* p = PE + (long long)t * DIM;
    p[2 * j]           = sinf(x * div);
    p[2 * j + 1]       = cosf(x * div);
    p[256 + 2 * j]     = sinf(y * div);
    p[256 + 2 * j + 1] = cosf(y * div);
}

// ---------------- H[0]=cls; H[r]=LN(G1[r-1])*g+b + G2[r-1] ----------------
__global__ void combine_embed(const float* __restrict__ G1, const float* __restrict__ G2,
                              const float* __restrict__ g, const float* __restrict__ b,
                              const float* __restrict__ cls, float* __restrict__ H)
{
    int row = blockIdx.x;               // 0..NTOK-1
    int tid = threadIdx.x;              // 256
    if (row == 0) {
        for (int c = tid; c < DIM; c += 256) H[c] = cls[c];
        return;
    }
    const float* src = G1 + (long long)(row - 1) * DIM;
    const float* pe  = G2 + (long long)(row - 1) * DIM;
    __shared__ float red[256];
    float v0 = src[tid], v1 = src[tid + 256];
    red[tid] = v0 + v1; __syncthreads();
    for (int s = 128; s > 0; s >>= 1) { if (tid < s) red[tid] += red[tid + s]; __syncthreads(); }
    float mu = red[0] / 512.0f; __syncthreads();
    float d0 = v0 - mu, d1 = v1 - mu;
    red[tid] = d0 * d0 + d1 * d1; __syncthreads();
    for (int s = 128; s > 0; s >>= 1) { if (tid < s) red[tid] += red[tid + s]; __syncthreads(); }
    float rstd = rsqrtf(red[0] / 512.0f + 1e-5f);
    float* dst = H + (long long)row * DIM;
    dst[tid]       = d0 * rstd * g[tid]       + b[tid]       + pe[tid];
    dst[tid + 256] = d1 * rstd * g[tid + 256] + b[tid + 256] + pe[tid + 256];
}

// ---------------- generic LayerNorm over 512 cols ----------------
__global__ void ln_rows(const float* __restrict__ in, float* __restrict__ out,
                        const float* __restrict__ g, const float* __restrict__ b)
{
    int row = blockIdx.x;
    int tid = threadIdx.x;              // 256
    const float* src = in + (long long)row * DIM;
    __shared__ float red[256];
    float v0 = src[tid], v1 = src[tid + 256];
    red[tid] = v0 + v1; __syncthreads();
    for (int s = 128; s > 0; s >>= 1) { if (tid < s) red[tid] += red[tid + s]; __syncthreads(); }
    float mu = red[0] / 512.0f; __syncthreads();
    float d0 = v0 - mu, d1 = v1 - mu;
    red[tid] = d0 * d0 + d1 * d1; __syncthreads();
    for (int s = 128; s > 0; s >>= 1) { if (tid < s) red[tid] += red[tid + s]; __syncthreads(); }
    float rstd = rsqrtf(red[0] / 512.0f + 1e-5f);
    float* dst = out + (long long)row * DIM;
    dst[tid]       = d0 * rstd * g[tid]       + b[tid];
    dst[tid + 256] = d1 * rstd * g[tid + 256] + b[tid + 256];
}

// ---------------- landmark means: QL (scaled), KL ----------------
__global__ void landmark_kernel(const float* __restrict__ QKV,
                                float* __restrict__ QL, float* __restrict__ KL)
{
    int d = threadIdx.x;                // 64
    int m = blockIdx.x;                 // 256
    int h = blockIdx.y;                 // 8
    const float* p = QKV + (long long)m * CHUNK * 1536 + h * HD + d;
    float sq = 0.0f, sk = 0.0f;
    for (int c = 0; c < CHUNK; ++c) {
        sq += p[(long long)c * 1536];
        sk += p[(long long)c * 1536 + 512];
    }
    QL[((long long)h * LM + m) * HD + d] = sq * (1.0f / CHUNK) * QSCALE;
    KL[((long long)h * LM + m) * HD + d] = sk * (1.0f / CHUNK);
}

// ---------------- in-place row softmax, row length L ----------------
__global__ void softmax_rows(float* __restrict__ S, int L)
{
    long long row = blockIdx.x;
    float* p = S + row * (long long)L;
    int tid = threadIdx.x;              // 256
    __shared__ float red[256];
    float mx = -3.4e38f;
    for (int i = tid; i < L; i += 256) mx = fmaxf(mx, p[i]);
    red[tid] = mx; __syncthreads();
    for (int s = 128; s > 0; s >>= 1) { if (tid < s) red[tid] = fmaxf(red[tid], red[tid + s]); __syncthreads(); }
    mx = red[0]; __syncthreads();
    float sum = 0.0f;
    for (int i = tid; i < L; i += 256) { float e = expf(p[i] - mx); p[i] = e; sum += e; }
    red[tid] = sum; __syncthreads();
    for (int s = 128; s > 0; s >>= 1) { if (tid < s) red[tid] += red[tid + s]; __syncthreads(); }
    float inv = 1.0f / red[0];
    for (int i = tid; i < L; i += 256) p[i] *= inv;
}

// ---------------- pinv init: AK = A^T / (n1 * ninf) per head ----------------
__global__ void pinv_init(const float* __restrict__ K2, float* __restrict__ AK)
{
    int h = blockIdx.x;
    int j = threadIdx.x;                // 256
    const float* A = K2 + (long long)h * LM * LM;
    float* O = AK + (long long)h * LM * LM;
    float cs = 0.0f, rs = 0.0f;
    for (int i = 0; i < LM; ++i) {
        cs += fabsf(A[i * LM + j]);     // column sum (ord 1)
        rs += fabsf(A[j * LM + i]);     // row sum (ord inf)
    }
    __shared__ float r1[256], r2[256];
    r1[j] = cs; r2[j] = rs; __syncthreads();
    for (int s = 128; s > 0; s >>= 1) {
        if (j < s) { r1[j] = fmaxf(r1[j], r1[j + s]); r2[j] = fmaxf(r2[j], r2[j + s]); }
        __syncthreads();
    }
    float scale = 1.0f / (r1[0] * r2[0]);
    for (int i = 0; i < LM; ++i) O[i * LM + j] = A[j * LM + i] * scale;
}

// ---------------- T = 2I - T (in place), batched 256x256 ----------------
__global__ void two_i_minus(float* __restrict__ T)
{
    long long idx = (long long)blockIdx.x * blockDim.x + threadIdx.x;
    if (idx >= (long long)HEADS * LM * LM) return;
    int i = (int)((idx >> 8) & 255);
    int j = (int)(idx & 255);
    T[idx] = ((i == j) ? 2.0f : 0.0f) - T[idx];
}

// ---------------- final: LN(H[0]) -> gelu(.@Wc1+bc1)@Wc2+bc2 ----------------
__global__ void cls_head(const float* __restrict__ H,
                         const float* __restrict__ gf, const float* __restrict__ bf,
                         const float* __restrict__ Wc1, const float* __restrict__ bc1,
                         const float* __restrict__ Wc2, const float* __restrict__ bc2,
                         float* __restrict__ out)
{
    __shared__ float c[512];
    __shared__ float red[256];
    __shared__ float hid[256];
    int tid = threadIdx.x;              // 256
    float v0 = H[tid], v1 = H[tid + 256];
    red[tid] = v0 + v1; __syncthreads();
    for (int s = 128; s > 0; s >>= 1) { if (tid < s) red[tid] += red[tid + s]; __syncthreads(); }
    float mu = red[0] / 512.0f; __syncthreads();
    float d0 = v0 - mu, d1 = v1 - mu;
    red[tid] = d0 * d0 + d1 * d1; __syncthreads();
    for (int s = 128; s > 0; s >>= 1) { if (tid < s) red[tid] += red[tid + s]; __syncthreads(); }
    float rstd = rsqrtf(red[0] / 512.0f + 1e-5f);
    c[tid]       = d0 * rstd * gf[tid]       + bf[tid];
    c[tid + 256] = d1 * rstd * gf[tid + 256] + bf[tid + 256];
    __syncthreads();
    float acc = bc1[tid];
    for (int d = 0; d < 512; ++d) acc += c[d] * Wc1[d * 256 + tid];
    acc = 0.5f * acc * (1.0f + erff(acc * 0.7071067811865475f));
    hid[tid] = acc; __syncthreads();
    if (tid < 2) {
        float l = bc2[tid];
        for (int j = 0; j < 256; ++j) l += hid[j] * Wc2[j * 2 + tid];
        out[tid] = l;
    }
}

// =====================================================================
// host orchestration
// =====================================================================
static inline void gemm(hipStream_t s, const float* A, const float* B, float* C,
                        int M, int N, int K,
                        long long as_m, long long as_k, long long ab,
                        long long bs_k, long long bs_n, long long bb,
                        long long cs_m, long long cb,
                        const float* bias, const float* res, long long rs_m,
                        float alpha, int act, int batch)
{
    dim3 g((N + 63) / 64, (M + 63) / 64, batch);
    gemm_wmma_f32<<<g, dim3(128), 0, s>>>(A, B, C, M, N, K,
                                          as_m, as_k, ab, bs_k, bs_n, bb,
                                          cs_m, cb, bias, res, rs_m, alpha, act);
}

extern "C" void kernel_launch(void* const* d_in, const int* in_sizes, int n_in,
                              void* d_out, int out_size, void* d_ws, size_t ws_size,
                              hipStream_t stream)
{
    (void)in_sizes; (void)n_in; (void)out_size; (void)ws_size;

    // ---- inputs (setup_inputs dict insertion order, recursive) ----
    const float* x      = (const float*)d_in[0];
    const int*   coords = (const int*)  d_in[1];
    const float* W_in   = (const float*)d_in[2];
    const float* b_in   = (const float*)d_in[3];
    const float* lnin_g = (const float*)d_in[4];
    const float* lnin_b = (const float*)d_in[5];
    const float* W_pos  = (const float*)d_in[6];
    const float* b_pos  = (const float*)d_in[7];
    const float* cls    = (const float*)d_in[8];
    // layers at 9 + l*11: ln1_g,ln1_b,Wqkv,Wo,bo,ln2_g,ln2_b,W1,b1,W2,b2
    const float* lnf_g  = (const float*)d_in[31];
    const float* lnf_b  = (const float*)d_in[32];
    const float* Wc1    = (const float*)d_in[33];
    const float* bc1    = (const float*)d_in[34];
    const float* Wc2    = (const float*)d_in[35];
    const float* bc2    = (const float*)d_in[36];
    float* out = (float*)d_out;

    // ---- workspace carve-up (floats) ----
    float* ws   = (float*)d_ws;
    float* H    = ws;                                   // 20480*512
    float* SCR  = H   + (long long)NTOK * DIM;          // 20480*512 (XLN / ATT / LN2)
    float* QKV  = SCR + (long long)NTOK * DIM;          // 20480*1536
    float* BIG  = QKV + (long long)NTOK * 1536;         // 8*20480*256 (S3 -> S1 -> MLP hidden)
    float* QL   = BIG + (long long)HEADS * NTOK * LM;   // 8*256*64
    float* KL   = QL  + (long long)HEADS * LM * HD;
    float* K2   = KL  + (long long)HEADS * LM * HD;     // 8*256*256
    float* AKA  = K2  + (long long)HEADS * LM * LM;
    float* AKB  = AKA + (long long)HEADS * LM * LM;
    float* T1   = AKB + (long long)HEADS * LM * LM;
    float* Tm   = T1  + (long long)HEADS * LM * LM;     // 8*256*64
    float* Um   = Tm  + (long long)HEADS * LM * HD;     // 8*256*64
    // prologue aliases inside QKV region:
    float* G1 = QKV;
    float* PE = QKV + (long long)NTOK * DIM;
    float* G2 = QKV + 2LL * NTOK * DIM;

    const long long LL_LM2 = (long long)LM * LM;

    // ---- prologue ----
    pe_kernel<<<((long long)NT * 128 + 255) / 256, 256, 0, stream>>>(coords, PE);
    // G1 = x @ W_in + b_in
    gemm(stream, x, W_in, G1, NT, DIM, INDIM, INDIM, 1, 0, DIM, 1, 0, DIM, 0,
         b_in, nullptr, 0, 1.0f, 0, 1);
    // G2 = PE @ W_pos + b_pos
    gemm(stream, PE, W_pos, G2, NT, DIM, DIM, DIM, 1, 0, DIM, 1, 0, DIM, 0,
         b_pos, nullptr, 0, 1.0f, 0, 1);
    combine_embed<<<NTOK, 256, 0, stream>>>(G1, G2, lnin_g, lnin_b, cls, H);

    // ---- transformer layers ----
    for (int l = 0; l < 2; ++l) {
        const float* ln1_g = (const float*)d_in[9 + l * 11 + 0];
        const float* ln1_b = (const float*)d_in[9 + l * 11 + 1];
        const float* Wqkv  = (const float*)d_in[9 + l * 11 + 2];
        const float* Wo    = (const float*)d_in[9 + l * 11 + 3];
        const float* bo    = (const float*)d_in[9 + l * 11 + 4];
        const float* ln2_g = (const float*)d_in[9 + l * 11 + 5];
        const float* ln2_b = (const float*)d_in[9 + l * 11 + 6];
        const float* W1    = (const float*)d_in[9 + l * 11 + 7];
        const float* b1    = (const float*)d_in[9 + l * 11 + 8];
        const float* W2    = (const float*)d_in[9 + l * 11 + 9];
        const float* b2    = (const float*)d_in[9 + l * 11 + 10];

        // XLN = LN1(H)
        ln_rows<<<NTOK, 256, 0, stream>>>(H, SCR, ln1_g, ln1_b);
        // QKV = XLN @ Wqkv
        gemm(stream, SCR, Wqkv, QKV, NTOK, 1536, DIM, DIM, 1, 0, 1536, 1, 0, 1536, 0,
             nullptr, nullptr, 0, 1.0f, 0, 1);
        // landmarks
        landmark_kernel<<<dim3(LM, HEADS), HD, 0, stream>>>(QKV, QL, KL);
        // K2 = softmax(QL @ KL^T)  [8,256,256]
        gemm(stream, QL, KL, K2, LM, LM, HD, HD, 1, (long long)LM * HD,
             1, HD, (long long)LM * HD, LM, LL_LM2, nullptr, nullptr, 0, 1.0f, 0, HEADS);
        softmax_rows<<<HEADS * LM, 256, 0, stream>>>(K2, LM);
        // Newton-Schulz pinv
        pinv_init<<<HEADS, 256, 0, stream>>>(K2, AKA);
        float* cur = AKA; float* nxt = AKB;
        for (int it = 0; it < 6; ++it) {
            gemm(stream, K2, cur, T1, LM, LM, LM, LM, 1, LL_LM2, LM, 1, LL_LM2,
                 LM, LL_LM2, nullptr, nullptr, 0, 1.0f, 0, HEADS);
            two_i_minus<<<(HEADS * LM * LM + 255) / 256, 256, 0, stream>>>(T1);
            gemm(stream, cur, T1, nxt, LM, LM, LM, LM, 1, LL_LM2, LM, 1, LL_LM2,
                 LM, LL_LM2, nullptr, nullptr, 0, 1.0f, 0, HEADS);
            float* tmp = cur; cur = nxt; nxt = tmp;
        }
        // S3 = softmax(QL @ k^T)  [8,256,20480] into BIG
        gemm(stream, QL, QKV + 512, BIG, LM, NTOK, HD,
             HD, 1, (long long)LM * HD,          // A = QL
             1, 1536, 64,                        // B[k][n] = QKV[512 + h*64 + k + n*1536]
             NTOK, (long long)LM * NTOK, nullptr, nullptr, 0, 1.0f, 0, HEADS);
        softmax_rows<<<HEADS * LM, 256, 0, stream>>>(BIG, NTOK);
        // Tm = S3 @ v  [8,256,64]
        gemm(stream, BIG, QKV + 1024, Tm, LM, HD, NTOK,
             NTOK, 1, (long long)LM * NTOK,
             1536, 1, 64,
             HD, (long long)LM * HD, nullptr, nullptr, 0, 1.0f, 0, HEADS);
        // Um = pinv @ Tm
        gemm(stream, cur, Tm, Um, LM, HD, LM,
             LM, 1, LL_LM2, HD, 1, (long long)LM * HD,
             HD, (long long)LM * HD, nullptr, nullptr, 0, 1.0f, 0, HEADS);
        // S1 = softmax(scale * q @ KL^T)  [8,20480,256] into BIG (reuse)
        gemm(stream, QKV, KL, BIG, NTOK, LM, HD,
             1536, 1, 64,                        // A = q (head offset 64)
             1, HD, (long long)LM * HD,          // B = KL^T
             LM, (long long)NTOK * LM, nullptr, nullptr, 0, QSCALE, 0, HEADS);
        softmax_rows<<<HEADS * NTOK, 256, 0, stream>>>(BIG, LM);
        // ATT(SCR) = S1 @ Um  (per-head 64 columns of [20480,512])
        gemm(stream, BIG, Um, SCR, NTOK, HD, LM,
             LM, 1, (long long)NTOK * LM,
             HD, 1, (long long)LM * HD,
             DIM, 64, nullptr, nullptr, 0, 1.0f, 0, HEADS);
        // H = H + ATT @ Wo + bo
        gemm(stream, SCR, Wo, H, NTOK, DIM, DIM, DIM, 1, 0, DIM, 1, 0, DIM, 0,
             bo, H, DIM, 1.0f, 0, 1);
        // MLP: Z = LN2(H); BIG = gelu(Z@W1+b1); H += BIG@W2+b2
        ln_rows<<<NTOK, 256, 0, stream>>>(H, SCR, ln2_g, ln2_b);
        gemm(stream, SCR, W1, BIG, NTOK, MLPD, DIM, DIM, 1, 0, MLPD, 1, 0, MLPD, 0,
             b1, nullptr, 0, 1.0f, 1, 1);
        gemm(stream, BIG, W2, H, NTOK, DIM, MLPD, MLPD, 1, 0, DIM, 1, 0, DIM, 0,
             b2, H, DIM, 1.0f, 0, 1);
    }

    // ---- classifier head on CLS row ----
    cls_head<<<1, 256, 0, stream>>>(H, lnf_g, lnf_b, Wc1, bc1, Wc2, bc2, out);
}